// LQR_16655883174570
// MI455X (gfx1250) — compile-verified
//
#include <hip/hip_runtime.h>
#include <math.h>

// ---------------------------------------------------------------------------
// LQR (Riccati backward recursion + forward rollout) for MI455X / gfx1250.
// BATCH=16, T=512, n=64 (state), m=32 (ctrl), n+m=96.
// One persistent workgroup per batch; V held in LDS across all 512 steps.
// All GEMMs use V_WMMA_F32_16X16X4_F32 (the fp32 WMMA shape on CDNA5).
// ---------------------------------------------------------------------------

typedef __attribute__((ext_vector_type(2))) float v2f;
typedef __attribute__((ext_vector_type(8))) float v8f;

#define LQR_BATCH 16
#define LQR_T     512
#define NS 64          // n_state
#define NC 32          // n_ctrl
#define NF 96          // n+m

// LDS pitches (odd => conflict-free column-strided access for WMMA A frags)
#define VP 65
#define FP 97
#define QP 97
#define KP 65
#define MP 65
#define LP 33

// Accumulate a 16x16 f32 tile: acc += X(16 x Kdim) * Y(Kdim x 16).
// X element (mm,kk) at X[mm*xm + kk*xk]; Y element (kk,nn) at Y[kk*yk + nn*yn].
// Caller pre-offsets X to (row0,0) and Y to (0,col0).
// Fragment layout per CDNA5 ISA (32-bit A 16x4 / B 4x16):
//   lanes 0-15 carry K=0,1 ; lanes 16-31 carry K=2,3.
__device__ __forceinline__ v8f wmma_acc_f32(const float* X, int xm, int xk,
                                            const float* Y, int yk, int yn,
                                            int Kdim, v8f acc, int l, int half)
{
  #pragma unroll 4
  for (int kk = 0; kk < Kdim; kk += 4) {
    const int k0 = kk + 2 * half;
    v2f a, b;
    a.x = X[l * xm + (k0 + 0) * xk];
    a.y = X[l * xm + (k0 + 1) * xk];
    b.x = Y[(k0 + 0) * yk + l * yn];
    b.y = Y[(k0 + 1) * yk + l * yn];
    acc = __builtin_amdgcn_wmma_f32_16x16x4_f32(false, a, false, b,
                                                (short)0, acc, false, false);
  }
  return acc;
}

// C/D tile layout: VGPR r holds row (r + 8*half), lanes 0-15 -> cols 0-15.
__device__ __forceinline__ v8f load_tile(const float* S, int pitch, int l, int half)
{
  v8f c;
  #pragma unroll
  for (int r = 0; r < 8; ++r) c[r] = S[(r + 8 * half) * pitch + l];
  return c;
}

__device__ __forceinline__ void store_tile(float* S, int pitch, int l, int half, v8f c)
{
  #pragma unroll
  for (int r = 0; r < 8; ++r) S[(r + 8 * half) * pitch + l] = c[r];
}

// ---------------------------------------------------------------------------
// Backward Riccati pass: one block (256 threads = 8 waves) per batch.
// Produces K (32x64) and k (32) per (b,t) into workspace.
// ---------------------------------------------------------------------------
extern "C" __global__ void __launch_bounds__(256, 1)
lqr_backward(const float* __restrict__ Q, const float* __restrict__ p,
             const float* __restrict__ A, const float* __restrict__ Bm,
             const float* __restrict__ c1, float* __restrict__ Kout,
             float* __restrict__ kout)
{
  __shared__ float sV[NS * VP];     // value matrix V (persists across steps)
  __shared__ float sF[NS * FP];     // F = [A B]  (64 x 96)
  __shared__ float sQ[NF * QP];     // Qh = Q + F^T V F (96 x 96)
  __shared__ float sFtV[NF * VP];   // F^T V (96 x 64)
  __shared__ float sK[NC * KP];     // K (32 x 64)
  __shared__ float sM[NC * MP];     // M1 = Qux + Quu K (32 x 64)
  __shared__ float sL[NC * LP];     // Cholesky factor of Quu (32 x 32)
  __shared__ float sv[NS];          // value vector v
  __shared__ float sqh[NF];         // qh
  __shared__ float st1[NS];         // V c + v
  __shared__ float st2[NC];         // qu + Quu k
  __shared__ float sk[NC];          // k

  const int tid  = threadIdx.x;
  const int wave = tid >> 5;
  const int lane = tid & 31;
  const int l    = lane & 15;
  const int half = lane >> 4;
  const int b    = blockIdx.x;

  for (int i = tid; i < NS * VP; i += 256) sV[i] = 0.0f;
  if (tid < NS) sv[tid] = 0.0f;
  __syncthreads();

  for (int t = LQR_T - 1; t >= 0; --t) {
    const size_t bt = (size_t)b * LQR_T + t;
    const float* Qg = Q  + bt * (NF * NF);
    const float* pg = p  + bt * NF;
    const float* Ag = A  + bt * (NS * NS);
    const float* Bg = Bm + bt * (NS * NC);
    const float* cg = c1 + bt * NS;

    // Warm L2 for next step (global_prefetch_b8).
    if (t > 0) {
      const float* Qn = Qg - NF * NF;
      const float* An = Ag - NS * NS;
      for (int i = tid; i < NF * NF / 16; i += 256) __builtin_prefetch(Qn + i * 16, 0, 1);
      for (int i = tid; i < NS * NS / 16; i += 256) __builtin_prefetch(An + i * 16, 0, 1);
    }

    // Load F = [A | B] into LDS (64 x 96).
    for (int i = tid; i < NS * NS; i += 256) {
      int r = i >> 6, c = i & 63;
      sF[r * FP + c] = Ag[i];
    }
    for (int i = tid; i < NS * NC; i += 256) {
      int r = i >> 5, c = i & 31;
      sF[r * FP + 64 + c] = Bg[i];
    }
    __syncthreads();

    // FtV = F^T (96x64) * V (64x64).  X = F^T: (m,k) -> sF[k*FP + m].
    for (int tile = wave; tile < 24; tile += 8) {
      const int tr = (tile >> 2) * 16, tc = (tile & 3) * 16;
      v8f acc = {};
      acc = wmma_acc_f32(sF + tr, 1, FP, sV + tc, VP, 1, NS, acc, l, half);
      store_tile(sFtV + tr * VP + tc, VP, l, half, acc);
    }
    __syncthreads();

    // Qh = Q + FtV (96x64) @ F (64x96); C initialized straight from global Q.
    for (int tile = wave; tile < 36; tile += 8) {
      const int tr = (tile / 6) * 16, tc = (tile % 6) * 16;
      v8f acc = load_tile(Qg + tr * NF + tc, NF, l, half);
      acc = wmma_acc_f32(sFtV + tr * VP, VP, 1, sF + tc, FP, 1, NS, acc, l, half);
      store_tile(sQ + tr * QP + tc, QP, l, half, acc);
    }
    __syncthreads();

    // t1 = V c + v
    if (tid < NS) {
      float acc = sv[tid];
      for (int j = 0; j < NS; ++j) acc += sV[tid * VP + j] * cg[j];
      st1[tid] = acc;
    }
    __syncthreads();

    // qh = p + F^T t1 ; copy Quu into sL for in-place Cholesky.
    if (tid < NF) {
      float acc = pg[tid];
      for (int i = 0; i < NS; ++i) acc += sF[i * FP + tid] * st1[i];
      sqh[tid] = acc;
    }
    for (int i = tid; i < NC * NC; i += 256) {
      int r = i >> 5, c = i & 31;
      sL[r * LP + c] = sQ[(64 + r) * QP + 64 + c];
    }
    __syncthreads();

    // Cholesky of Quu (wave 0 only; lockstep LDS column updates).
    if (wave == 0) {
      const int i = lane;
      for (int j = 0; j < NC; ++j) {
        const float d = sqrtf(sL[j * LP + j]);
        float lij;
        if (i == j)      lij = d;
        else if (i > j)  lij = sL[i * LP + j] / d;
        else             lij = 0.0f;
        sL[i * LP + j] = lij;
        for (int k2 = j + 1; k2 < NC; ++k2)
          if (i >= k2) sL[i * LP + k2] -= lij * sL[k2 * LP + j];
      }
    }
    __syncthreads();

    // Solve Quu X = [Qux | qu]; K = -X[:, :64], k = -X[:, 64].
    // One thread per RHS column; column kept in registers (fully unrolled).
    if (tid < 65) {
      float y[NC];
      #pragma unroll
      for (int i = 0; i < NC; ++i) {
        float r = (tid < 64) ? sQ[(64 + i) * QP + tid] : sqh[64 + i];
        #pragma unroll
        for (int j = 0; j < NC; ++j)
          if (j < i) r -= sL[i * LP + j] * y[j];
        y[i] = r / sL[i * LP + i];
      }
      #pragma unroll
      for (int i = NC - 1; i >= 0; --i) {
        float r = y[i];
        #pragma unroll
        for (int j = 0; j < NC; ++j)
          if (j > i) r -= sL[j * LP + i] * y[j];
        y[i] = r / sL[i * LP + i];
      }
      if (tid < 64) {
        #pragma unroll
        for (int i = 0; i < NC; ++i) sK[i * KP + tid] = -y[i];
      } else {
        #pragma unroll
        for (int i = 0; i < NC; ++i) sk[i] = -y[i];
      }
    }
    __syncthreads();

    // Persist K, k for the forward pass.
    float* Kg = Kout + bt * (NC * NS);
    for (int i = tid; i < NC * NS; i += 256) {
      int r = i >> 6, c = i & 63;
      Kg[i] = sK[r * KP + c];
    }
    if (tid < NC) kout[bt * NC + tid] = sk[tid];

    // M1 = Qux + Quu @ K  (32x64); one 16x16 tile per wave.
    for (int tile = wave; tile < 8; tile += 8) {
      const int tr = (tile >> 2) * 16, tc = (tile & 3) * 16;
      v8f acc = load_tile(sQ + (64 + tr) * QP + tc, QP, l, half);      // Qux
      acc = wmma_acc_f32(sQ + (64 + tr) * QP + 64, QP, 1,              // Quu
                         sK + tc, KP, 1, NC, acc, l, half);
      store_tile(sM + tr * MP + tc, MP, l, half, acc);
    }
    __syncthreads();

    // Vn = Qxx + Qxu @ K + K^T @ M1  (64x64) -> overwrite sV.
    for (int tile = wave; tile < 16; tile += 8) {
      const int tr = (tile >> 2) * 16, tc = (tile & 3) * 16;
      v8f acc = load_tile(sQ + tr * QP + tc, QP, l, half);             // Qxx
      acc = wmma_acc_f32(sQ + tr * QP + 64, QP, 1,                     // Qxu @ K
                         sK + tc, KP, 1, NC, acc, l, half);
      acc = wmma_acc_f32(sK + tr, 1, KP,                               // K^T @ M1
                         sM + tc, MP, 1, NC, acc, l, half);
      store_tile(sV + tr * VP + tc, VP, l, half, acc);
    }
    // t2 = qu + Quu k
    if (tid < NC) {
      float acc = sqh[64 + tid];
      for (int j = 0; j < NC; ++j) acc += sQ[(64 + tid) * QP + 64 + j] * sk[j];
      st2[tid] = acc;
    }
    __syncthreads();

    // vn = qx + Qxu k + K^T t2 ; symmetrize V in place.
    if (tid < NS) {
      float acc = sqh[tid];
      for (int j = 0; j < NC; ++j) acc += sQ[tid * QP + 64 + j] * sk[j];
      for (int j = 0; j < NC; ++j) acc += sK[j * KP + tid] * st2[j];
      sv[tid] = acc;
    }
    for (int i = tid; i < NS * NS; i += 256) {
      int r = i >> 6, c = i & 63;
      if (r < c) {
        float m0 = 0.5f * (sV[r * VP + c] + sV[c * VP + r]);
        sV[r * VP + c] = m0;
        sV[c * VP + r] = m0;
      }
    }
    __syncthreads();
  }
}

// ---------------------------------------------------------------------------
// Forward rollout: x_{t+1} = A x + B u + c, u = K x + k ; tau = [x, u].
// One block (96 threads) per batch.
// ---------------------------------------------------------------------------
extern "C" __global__ void __launch_bounds__(96, 1)
lqr_forward(const float* __restrict__ A, const float* __restrict__ Bm,
            const float* __restrict__ c1, const float* __restrict__ x0,
            const float* __restrict__ Kin, const float* __restrict__ kin,
            float* __restrict__ out)
{
  __shared__ float sx[NS];
  __shared__ float su[NC];
  __shared__ float sxn[NS];
  const int tid = threadIdx.x;
  const int b   = blockIdx.x;

  if (tid < NS) sx[tid] = x0[b * NS + tid];
  __syncthreads();

  for (int t = 0; t < LQR_T; ++t) {
    const size_t bt = (size_t)b * LQR_T + t;
    const float* Kg = Kin + bt * (NC * NS);
    const float* Ag = A   + bt * (NS * NS);
    const float* Bg = Bm  + bt * (NS * NC);
    float*       og = out + bt * NF;

    if (tid >= NS) {                       // u = K x + k   (threads 64..95)
      const int j = tid - NS;
      float acc = kin[bt * NC + j];
      for (int i = 0; i < NS; ++i) acc += Kg[j * NS + i] * sx[i];
      su[j] = acc;
    }
    __syncthreads();

    og[tid] = (tid < NS) ? sx[tid] : su[tid - NS];   // tau[b,t] = [x, u]

    if (tid < NS) {                        // x' = A x + B u + c
      float acc = c1[bt * NS + tid];
      for (int j = 0; j < NS; ++j) acc += Ag[tid * NS + j] * sx[j];
      for (int j = 0; j < NC; ++j) acc += Bg[tid * NC + j] * su[j];
      sxn[tid] = acc;
    }
    __syncthreads();
    if (tid < NS) sx[tid] = sxn[tid];
    __syncthreads();
  }
}

// ---------------------------------------------------------------------------
// Host-side launch. Inputs (dict order): Q, p, A, B, c1, x_init.
// Output: tau (16, 512, 96) float32. Workspace: K (16*512*32*64) + k
// (16*512*32) floats = ~68.2 MB.
// ---------------------------------------------------------------------------
extern "C" void kernel_launch(void* const* d_in, const int* in_sizes, int n_in,
                              void* d_out, int out_size, void* d_ws, size_t ws_size,
                              hipStream_t stream)
{
  (void)in_sizes; (void)n_in; (void)out_size; (void)ws_size;

  const float* Q  = (const float*)d_in[0];
  const float* p  = (const float*)d_in[1];
  const float* A  = (const float*)d_in[2];
  const float* Bm = (const float*)d_in[3];
  const float* c1 = (const float*)d_in[4];
  const float* x0 = (const float*)d_in[5];
  float* out = (float*)d_out;

  float* Kws = (float*)d_ws;                                   // 16*512*32*64
  float* kws = Kws + (size_t)LQR_BATCH * LQR_T * NC * NS;      // 16*512*32

  lqr_backward<<<LQR_BATCH, 256, 0, stream>>>(Q, p, A, Bm, c1, Kws, kws);
  lqr_forward<<<LQR_BATCH, 96, 0, stream>>>(A, Bm, c1, x0, Kws, kws, out);
}